// AffineChamferLoss_9955734192761
// MI455X (gfx1250) — compile-verified
//
#include <hip/hip_runtime.h>

typedef float v2f __attribute__((ext_vector_type(2)));
typedef float v8f __attribute__((ext_vector_type(8)));

// ---------------------------------------------------------------------------
// Kernel 1: build WMMA operand forms.
//   row form (A side): (-2*p0, -2*p1, -2*p2, 1)
//   col form (B side): ( p0,    p1,    p2,  |p|^2)
// so that V_WMMA_F32_16X16X4_F32(A_i, B_j) = |q_j|^2 - 2 p_i.q_j, and the
// row term |p_i|^2 plus the max(.,0) clamp commute with the column-min and
// are applied once after the loop.  Moving points are affine-transformed.
// ---------------------------------------------------------------------------
__global__ __launch_bounds__(256) void chamfer_prep_kernel(
    const float* __restrict__ pts_fixed,   // [M,3]
    const float* __restrict__ pts_mov,     // [N,3]
    const float* __restrict__ mat,         // [3,3] row-major
    const float* __restrict__ trans,       // [3]
    float* __restrict__ xrow, float* __restrict__ xcol, float* __restrict__ x2,
    float* __restrict__ yrow, float* __restrict__ ycol, float* __restrict__ y2,
    int N, int M)
{
    int i = blockIdx.x * blockDim.x + threadIdx.x;
    if (i < N) {
        float p0 = pts_mov[3 * i + 0];
        float p1 = pts_mov[3 * i + 1];
        float p2 = pts_mov[3 * i + 2];
        float x[3];
        float s = 0.0f;
#pragma unroll
        for (int k = 0; k < 3; ++k) {
            x[k] = p0 * mat[0 * 3 + k] + p1 * mat[1 * 3 + k] +
                   p2 * mat[2 * 3 + k] + trans[k];
            s += x[k] * x[k];
        }
#pragma unroll
        for (int k = 0; k < 3; ++k) {
            xrow[4 * i + k] = -2.0f * x[k];
            xcol[4 * i + k] = x[k];
        }
        xrow[4 * i + 3] = 1.0f;
        xcol[4 * i + 3] = s;
        x2[i] = s;
    }
    if (i < M) {
        float q0 = pts_fixed[3 * i + 0];
        float q1 = pts_fixed[3 * i + 1];
        float q2 = pts_fixed[3 * i + 2];
        float s = q0 * q0 + q1 * q1 + q2 * q2;
        yrow[4 * i + 0] = -2.0f * q0;
        yrow[4 * i + 1] = -2.0f * q1;
        yrow[4 * i + 2] = -2.0f * q2;
        yrow[4 * i + 3] = 1.0f;
        ycol[4 * i + 0] = q0;
        ycol[4 * i + 1] = q1;
        ycol[4 * i + 2] = q2;
        ycol[4 * i + 3] = s;
        y2[i] = s;
    }
}

// ---------------------------------------------------------------------------
// Kernel 2: per 16-row tile (one wave32), min over all NQ columns of
// acc = |q|^2 - 2 p.q via one V_WMMA_F32_16X16X4_F32 per 16-col tile.
// Inner loop per tile: 1x global_load_b64 + 1x wmma + 8 running mins.
// A operand (16x4): lane holds row M=lane%16, K pair selected by lane half.
// B operand (4x16): VGPR i holds K-row i (+2 for upper lanes), lane%16 = col.
// C/D layout: VGPR v, lanes 0-15 -> row v, lanes 16-31 -> row v+8.
// ---------------------------------------------------------------------------
#define WAVES_PER_BLOCK 8

__global__ __launch_bounds__(256) void chamfer_min_kernel(
    const float* __restrict__ prow,   // [NP,4] row-form (A side)
    const float* __restrict__ p2,     // [NP] squared norms of rows
    const float* __restrict__ qcol,   // [NQ,4] col-form (B side)
    float* __restrict__ minp,         // [NP] out: min_j max(d2(i,j),0)
    int NP, int NQ)
{
    const int lane = threadIdx.x & 31;
    const int wave = threadIdx.x >> 5;
    const int rowbase = (blockIdx.x * WAVES_PER_BLOCK + wave) * 16;
    if (rowbase >= NP) return;  // wave-uniform: EXEC stays all-1s for WMMA

    const int half = lane >> 4;
    const int l16  = lane & 15;

    // A operand: two K-components of row (rowbase + l16)
    const v2f a = *(const v2f*)(prow + 4 * (rowbase + l16) + 2 * half);

    float rmin[8];
#pragma unroll
    for (int v = 0; v < 8; ++v) rmin[v] = 3.4e38f;

    const float* qp = qcol + 4 * l16 + 2 * half;  // per-lane B base

#pragma unroll 4
    for (int colbase = 0; colbase < NQ; colbase += 16) {
        const v2f b = *(const v2f*)(qp + 4 * colbase);

        v8f acc = {};
        acc = __builtin_amdgcn_wmma_f32_16x16x4_f32(
            /*neg_a=*/false, a, /*neg_b=*/false, b,
            /*c_mod=*/(short)0, acc, /*reuse_a=*/false, /*reuse_b=*/false);

#pragma unroll
        for (int v = 0; v < 8; ++v)
            rmin[v] = fminf(rmin[v], acc[v]);
    }

    // min across the 16 columns held by lanes within each half
#pragma unroll
    for (int m = 1; m <= 8; m <<= 1) {
#pragma unroll
        for (int v = 0; v < 8; ++v)
            rmin[v] = fminf(rmin[v], __shfl_xor(rmin[v], m, 32));
    }

    // apply deferred row term and clamp:  max(|p|^2 + min_j acc, 0)
    if (l16 == 0) {
#pragma unroll
        for (int v = 0; v < 8; ++v) {
            int row = rowbase + v + 8 * half;
            minp[row] = fmaxf(rmin[v] + p2[row], 0.0f);
        }
    }
}

// ---------------------------------------------------------------------------
// Kernel 3: out = mean(minx) + mean(miny)
// ---------------------------------------------------------------------------
__global__ __launch_bounds__(256) void chamfer_reduce_kernel(
    const float* __restrict__ minx, const float* __restrict__ miny,
    float* __restrict__ out, int N, int M)
{
    __shared__ float smem[256];
    float sx = 0.0f, sy = 0.0f;
    for (int i = threadIdx.x; i < N; i += 256) sx += minx[i];
    for (int i = threadIdx.x; i < M; i += 256) sy += miny[i];
    smem[threadIdx.x] = sx / (float)N + sy / (float)M;
    __syncthreads();
    for (int s = 128; s > 0; s >>= 1) {
        if (threadIdx.x < s) smem[threadIdx.x] += smem[threadIdx.x + s];
        __syncthreads();
    }
    if (threadIdx.x == 0) out[0] = smem[0];
}

// ---------------------------------------------------------------------------
extern "C" void kernel_launch(void* const* d_in, const int* in_sizes, int n_in,
                              void* d_out, int out_size, void* d_ws, size_t ws_size,
                              hipStream_t stream) {
    const float* pts_fixed = (const float*)d_in[0];  // [M,3]
    const float* pts_mov   = (const float*)d_in[1];  // [N,3]
    const float* mat       = (const float*)d_in[2];  // [1,3,3]
    const float* trans     = (const float*)d_in[3];  // [1,3,1]

    const int M = in_sizes[0] / 3;
    const int N = in_sizes[1] / 3;

    float* ws   = (float*)d_ws;
    float* xrow = ws;                      // 4N
    float* xcol = xrow + (size_t)4 * N;    // 4N
    float* yrow = xcol + (size_t)4 * N;    // 4M
    float* ycol = yrow + (size_t)4 * M;    // 4M
    float* x2   = ycol + (size_t)4 * M;    // N
    float* y2   = x2 + N;                  // M
    float* minx = y2 + M;                  // N
    float* miny = minx + N;                // M

    const int maxNM = (N > M) ? N : M;
    chamfer_prep_kernel<<<(maxNM + 255) / 256, 256, 0, stream>>>(
        pts_fixed, pts_mov, mat, trans, xrow, xcol, x2, yrow, ycol, y2, N, M);

    // moving -> fixed
    {
        int tiles  = (N + 15) / 16;
        int blocks = (tiles + WAVES_PER_BLOCK - 1) / WAVES_PER_BLOCK;
        chamfer_min_kernel<<<blocks, 32 * WAVES_PER_BLOCK, 0, stream>>>(
            xrow, x2, ycol, minx, N, M);
    }
    // fixed -> moving
    {
        int tiles  = (M + 15) / 16;
        int blocks = (tiles + WAVES_PER_BLOCK - 1) / WAVES_PER_BLOCK;
        chamfer_min_kernel<<<blocks, 32 * WAVES_PER_BLOCK, 0, stream>>>(
            yrow, y2, xcol, miny, M, N);
    }

    chamfer_reduce_kernel<<<1, 256, 0, stream>>>(minx, miny, (float*)d_out, N, M);
}